// EdgeTransformer_36249523978497
// MI455X (gfx1250) — compile-verified
//
#include <hip/hip_runtime.h>
#include <hip/hip_bf16.h>

// ---------------- problem constants ----------------
#define NROWS 65536
#define HID   256
#define NHEAD 8
#define DHEAD 32
#define FF    1024
#define EPSLN 1e-5f

// ---------------- GEMM tiling ----------------
#define TILE_ROWS  128       // rows per block (8 waves x 16)
#define PANEL_K    32        // K rows staged per LDS panel
#define LDS_STRIDE 272       // 256 + 16 dword pad (bank-conflict free half-wave reads)

typedef __attribute__((ext_vector_type(2)))  float v2f;
typedef __attribute__((ext_vector_type(8)))  float v8f;
typedef __attribute__((ext_vector_type(4)))  unsigned int v4u;
typedef __attribute__((ext_vector_type(8)))  int v8i;
typedef __attribute__((ext_vector_type(4)))  int v4i;

enum { EPI_BIAS = 0, EPI_BIAS_RES = 1, EPI_BIAS_GELU = 2 };

// wave32 butterfly sum (deterministic)
__device__ __forceinline__ float et_wsum(float v) {
#pragma unroll
  for (int o = 16; o > 0; o >>= 1) v += __shfl_xor(v, o, 32);
  return v;
}

// ---------------- Tensor Data Mover: load a PANEL_K x 256 f32 panel to LDS ----
// Builds a D# (groups 0/1; groups 2/3 zero => 2D tensor) and issues
// tensor_load_to_lds (6-arg clang-23 form). Padding: 16 DWORDs inserted every
// 256 DWORDs so the LDS row stride becomes LDS_STRIDE floats.
__device__ __forceinline__ void et_tdm_load_panel(const float* gsrc,
                                                  unsigned lds_byte_addr,
                                                  unsigned ldW /*elements*/) {
  unsigned long long ga = (unsigned long long)(size_t)gsrc;
  v4u g0;
  g0.x = 1u;                                   // count=1 (valid user descriptor)
  g0.y = lds_byte_addr;                        // LDS byte address
  g0.z = (unsigned)ga;                         // global_addr[31:0]
  g0.w = (unsigned)(ga >> 32) | 0x80000000u;   // global_addr[56:32] | type=2
  v8i g1;
  g1.s0 = (int)((2u << 16)      // data_size = 4B
              | (1u << 20)      // pad_enable
              | (7u << 22)      // pad_interval: every 256 DWORDs
              | (15u << 25));   // pad_amount: 16 DWORDs
  g1.s1 = (int)(256u << 16);    // tensor_dim0 = 256 (bits 48..79, low half)
  g1.s2 = (int)(32u << 16);     // dim0 hi=0 | tensor_dim1 = 32 (low half)
  g1.s3 = (int)(256u << 16);    // dim1 hi=0 | tile_dim0 = 256
  g1.s4 = (int)32;              // tile_dim1 = 32, tile_dim2 = 0
  g1.s5 = (int)ldW;             // tensor_dim0_stride[31:0]
  g1.s6 = 0;                    // stride hi / dim1_stride lo
  g1.s7 = 0;
  v4i z4 = {0, 0, 0, 0};
  v8i z8 = {0, 0, 0, 0, 0, 0, 0, 0};
  __builtin_amdgcn_tensor_load_to_lds(g0, g1, z4, z4, z8, 0);
}

// ---------------- WMMA f32 GEMM: out[rows x Nout] = A[rows x K] @ W[K x Nout] -
template <int EPI>
__global__ __launch_bounds__(256) void et_gemm_wmma(
    const float* __restrict__ A, const float* __restrict__ W,
    const float* __restrict__ bias, const float* __restrict__ res,
    float* __restrict__ out, int K, int Nout) {
  __shared__ float wlds[2][PANEL_K * LDS_STRIDE];

  const int tid  = threadIdx.x;
  const int wid  = tid >> 5;
  const int lane = tid & 31;
  const int l16  = lane & 15;
  const int hi   = lane >> 4;                    // half-wave id (K split)
  const int rowBase = blockIdx.x * TILE_ROWS + wid * 16;
  const int c0      = blockIdx.y * 256;
  const float* __restrict__ Arow = A + (size_t)(rowBase + l16) * K;
  const int P = K / PANEL_K;

  v8f acc[16];
  const v8f zacc = {0.f, 0.f, 0.f, 0.f, 0.f, 0.f, 0.f, 0.f};
#pragma unroll
  for (int j = 0; j < 16; ++j) acc[j] = zacc;

  if (wid == 0) {  // wave 0 drives the TDM (EXEC-independent, wave-level op)
    et_tdm_load_panel(W + c0, (unsigned)(size_t)&wlds[0][0], (unsigned)Nout);
  }

  for (int p = 0; p < P; ++p) {
    if (wid == 0) {
      if (p + 1 < P) {
        et_tdm_load_panel(W + (size_t)(p + 1) * PANEL_K * Nout + c0,
                          (unsigned)(size_t)&wlds[(p + 1) & 1][0],
                          (unsigned)Nout);
        __builtin_amdgcn_s_wait_tensorcnt(1);  // panel p landed; p+1 in flight
      } else {
        __builtin_amdgcn_s_wait_tensorcnt(0);
      }
    }
    __syncthreads();

    const float* __restrict__ lp = &wlds[p & 1][0];
    const int kbase = p * PANEL_K;
    for (int kk = 0; kk < PANEL_K; kk += 4) {
      // A fragment: lane l (row rowBase+l16) supplies K = kk + hi*2 .. +1
      const v2f a = *(const v2f*)(Arow + kbase + kk + hi * 2);
      const int r0 = (kk + hi * 2) * LDS_STRIDE + l16;
#pragma unroll
      for (int j = 0; j < 16; ++j) {
        v2f b;
        b.x = lp[r0 + j * 16];               // K = kk+hi*2,   col j*16+l16
        b.y = lp[r0 + LDS_STRIDE + j * 16];  // K = kk+hi*2+1
        acc[j] = __builtin_amdgcn_wmma_f32_16x16x4_f32(
            false, a, false, b, (short)0, acc[j], false, false);
      }
    }
    __syncthreads();  // everyone done with buf[p&1] before it is refilled
  }

  // epilogue: C/D layout — VGPR i holds row (hi*8 + i), col = l16 per tile j
#pragma unroll
  for (int j = 0; j < 16; ++j) {
    const int col = c0 + j * 16 + l16;
    const float bv = bias[col];
#pragma unroll
    for (int i = 0; i < 8; ++i) {
      const int r = rowBase + hi * 8 + i;
      float v = acc[j][i] + bv;
      if (EPI == EPI_BIAS_GELU)
        v = 0.5f * v * (1.0f + erff(v * 0.7071067811865476f));
      if (EPI == EPI_BIAS_RES) v += res[(size_t)r * Nout + col];
      out[(size_t)r * Nout + col] = v;
    }
  }
}

// ---------------- LayerNorm: wave-per-row, optional dual output --------------
template <bool DUAL>
__global__ __launch_bounds__(256) void et_ln_kernel(
    const float* __restrict__ h, const float* __restrict__ g1,
    const float* __restrict__ b1, const float* __restrict__ g2,
    const float* __restrict__ b2, float* __restrict__ o1,
    float* __restrict__ o2) {
  const int wid = threadIdx.x >> 5, lane = threadIdx.x & 31;
  const int row = blockIdx.x * 8 + wid;
  const float* __restrict__ hr = h + (size_t)row * HID;
  const int c0 = lane * 4;
  float x[8];
#pragma unroll
  for (int i = 0; i < 4; ++i) { x[i] = hr[c0 + i]; x[4 + i] = hr[128 + c0 + i]; }
  float s = 0.f;
#pragma unroll
  for (int i = 0; i < 8; ++i) s += x[i];
  const float m = et_wsum(s) * (1.0f / HID);
  float ss = 0.f;
#pragma unroll
  for (int i = 0; i < 8; ++i) { const float d = x[i] - m; ss += d * d; }
  const float rs = rsqrtf(et_wsum(ss) * (1.0f / HID) + EPSLN);
  float* __restrict__ o1r = o1 + (size_t)row * HID;
  float* __restrict__ o2r = DUAL ? (o2 + (size_t)row * HID) : nullptr;
#pragma unroll
  for (int i = 0; i < 8; ++i) {
    const int col = (i < 4) ? (c0 + i) : (128 + c0 + i - 4);
    const float d = (x[i] - m) * rs;
    o1r[col] = d * g1[col] + b1[col];
    if (DUAL) o2r[col] = d * g2[col] + b2[col];
  }
}

// ---------------- stats pass 1: per-block partials (deterministic) -----------
// part[b][0]=sum q^2, [1]=sum k^2, [2+j]=sum_n k*v (col j), [258+j]=sum_n k
__global__ __launch_bounds__(256) void et_stats_partial(
    const float* __restrict__ q, const float* __restrict__ k,
    const float* __restrict__ v, float* __restrict__ part) {
  const int j = threadIdx.x;
  float dkv = 0.f, ks = 0.f, qq = 0.f, k2 = 0.f;
  const size_t base0 = (size_t)blockIdx.x * 128 * HID + j;
  for (int r = 0; r < 128; ++r) {
    const size_t idx = base0 + (size_t)r * HID;
    const float qv = q[idx], kv = k[idx], vv = v[idx];
    dkv += kv * vv; ks += kv; qq += qv * qv; k2 += kv * kv;
  }
  float* pb = part + (size_t)blockIdx.x * 514;
  pb[2 + j]   = dkv;
  pb[258 + j] = ks;
  qq = et_wsum(qq); k2 = et_wsum(k2);
  __shared__ float sq[8], sk[8];
  const int wid = threadIdx.x >> 5, lane = threadIdx.x & 31;
  if (lane == 0) { sq[wid] = qq; sk[wid] = k2; }
  __syncthreads();
  if (threadIdx.x == 0) {
    float a = 0.f, b = 0.f;
    for (int w = 0; w < 8; ++w) { a += sq[w]; b += sk[w]; }
    pb[0] = a; pb[1] = b;
  }
}

// ---------------- stats pass 2: reduce 512 partials ---------------------------
__global__ void et_stats_reduce(const float* __restrict__ part,
                                float* __restrict__ stats) {
  const int i = blockIdx.x * 256 + threadIdx.x;
  if (i < 514) {
    float s = 0.f;
    for (int b = 0; b < 512; ++b) s += part[(size_t)b * 514 + i];
    stats[i] = s;
  }
}

// ---------------- attention epilogue: wave w == head w (D=32=wave32) ----------
__global__ __launch_bounds__(256) void et_attn(
    const float* __restrict__ q, const float* __restrict__ v,
    const float* __restrict__ stats, float* __restrict__ attn) {
  const int j = threadIdx.x;
  const float inv = rsqrtf(stats[0]) * rsqrtf(stats[1]);  // 1/(||q|| ||k||)
  const float c1  = stats[2 + j] * inv;                   // q_norm*diag_norm coeff
  const float ksj = stats[258 + j];
  const size_t base0 = (size_t)blockIdx.x * 32 * HID + j;
  for (int r = 0; r < 32; ++r) {
    const size_t idx = base0 + (size_t)r * HID;
    const float qv = q[idx], vv = v[idx];
    const float s = et_wsum(qv * ksj);            // sum over d within head
    const float den = s * inv + (float)NROWS;
    attn[idx] = (qv * c1 + vv * (float)NROWS) / den;
  }
}

// ---------------- host orchestration -----------------------------------------
extern "C" void kernel_launch(void* const* d_in, const int* in_sizes, int n_in,
                              void* d_out, int out_size, void* d_ws,
                              size_t ws_size, hipStream_t stream) {
  const float* x    = (const float*)d_in[0];
  const float* Wq   = (const float*)d_in[1];
  const float* bq   = (const float*)d_in[2];
  const float* Wk   = (const float*)d_in[3];
  const float* bk   = (const float*)d_in[4];
  const float* Wv   = (const float*)d_in[5];
  const float* bv   = (const float*)d_in[6];
  const float* Wh   = (const float*)d_in[7];
  const float* bh   = (const float*)d_in[8];
  const float* g1kv = (const float*)d_in[9];
  const float* b1kv = (const float*)d_in[10];
  const float* g1q  = (const float*)d_in[11];
  const float* b1q  = (const float*)d_in[12];
  const float* Wf1  = (const float*)d_in[13];
  const float* bf1  = (const float*)d_in[14];
  const float* Wf2  = (const float*)d_in[15];
  const float* bf2  = (const float*)d_in[16];
  const float* g2   = (const float*)d_in[17];
  const float* b2   = (const float*)d_in[18];
  float* out = (float*)d_out;

  float* f = (float*)d_ws;
  const size_t NH = (size_t)NROWS * HID;          // 16.7M floats
  float* h_buf = f;                               // running hidden state
  float* lnkv  = f + 1 * NH;                      // also reused as attn buffer
  float* lnq   = f + 2 * NH;                      // also reused as post-LN buffer
  float* qb    = f + 3 * NH;
  float* kb    = f + 4 * NH;
  float* vb    = f + 5 * NH;
  float* tb    = f + 6 * NH;                      // FFN hidden: N x 1024 (4*NH)
  float* stats = f + 10 * NH;                     // 514 floats
  float* part  = stats + 1024;                    // 512 x 514 partials
  float* attn  = lnkv;
  float* lnp   = lnq;

  const dim3 blk(256);
  const dim3 gLN(NROWS / 8);
  const dim3 gGEMM(NROWS / TILE_ROWS, 1);
  const dim3 gGEMM4(NROWS / TILE_ROWS, 4);

  for (int l = 0; l < 3; ++l) {
    const float* hcur = (l == 0) ? x : h_buf;
    const float* Wq_l = Wq + (size_t)l * HID * HID;
    const float* Wk_l = Wk + (size_t)l * HID * HID;
    const float* Wv_l = Wv + (size_t)l * HID * HID;
    const float* Wh_l = Wh + (size_t)l * HID * HID;
    const float* Wf1_l = Wf1 + (size_t)l * HID * FF;
    const float* Wf2_l = Wf2 + (size_t)l * FF * HID;
    const float* bq_l = bq + (size_t)l * HID;
    const float* bk_l = bk + (size_t)l * HID;
    const float* bv_l = bv + (size_t)l * HID;
    const float* bh_l = bh + (size_t)l * HID;
    const float* bf1_l = bf1 + (size_t)l * FF;
    const float* bf2_l = bf2 + (size_t)l * HID;
    const float* g1kv_l = g1kv + (size_t)l * HID;
    const float* b1kv_l = b1kv + (size_t)l * HID;
    const float* g1q_l  = g1q  + (size_t)l * HID;
    const float* b1q_l  = b1q  + (size_t)l * HID;
    const float* g2_l   = g2   + (size_t)l * HID;
    const float* b2_l   = b2   + (size_t)l * HID;

    et_ln_kernel<true><<<gLN, blk, 0, stream>>>(hcur, g1kv_l, b1kv_l, g1q_l,
                                                b1q_l, lnkv, lnq);
    et_gemm_wmma<EPI_BIAS><<<gGEMM, blk, 0, stream>>>(lnq, Wq_l, bq_l, nullptr,
                                                      qb, HID, HID);
    et_gemm_wmma<EPI_BIAS><<<gGEMM, blk, 0, stream>>>(lnkv, Wk_l, bk_l, nullptr,
                                                      kb, HID, HID);
    et_gemm_wmma<EPI_BIAS><<<gGEMM, blk, 0, stream>>>(lnkv, Wv_l, bv_l, nullptr,
                                                      vb, HID, HID);
    et_stats_partial<<<dim3(NROWS / 128), blk, 0, stream>>>(qb, kb, vb, part);
    et_stats_reduce<<<dim3(3), blk, 0, stream>>>(part, stats);
    et_attn<<<dim3(NROWS / 32), blk, 0, stream>>>(qb, vb, stats, attn);
    et_gemm_wmma<EPI_BIAS_RES><<<gGEMM, blk, 0, stream>>>(attn, Wh_l, bh_l,
                                                          hcur, h_buf, HID, HID);
    et_ln_kernel<false><<<gLN, blk, 0, stream>>>(h_buf, g2_l, b2_l, nullptr,
                                                 nullptr, lnp, nullptr);
    et_gemm_wmma<EPI_BIAS_GELU><<<gGEMM4, blk, 0, stream>>>(lnp, Wf1_l, bf1_l,
                                                            nullptr, tb, HID, FF);
    et_gemm_wmma<EPI_BIAS_RES><<<gGEMM, blk, 0, stream>>>(
        tb, Wf2_l, bf2_l, h_buf, (l == 2) ? out : h_buf, FF, HID);
  }
}